// VGG16WithSCLFirst_7095285973802
// MI455X (gfx1250) — compile-verified
//
#include <hip/hip_runtime.h>
#include <hip/hip_bf16.h>

typedef __attribute__((ext_vector_type(2))) float v2f;
typedef __attribute__((ext_vector_type(8))) float v8f;

#define BN_EPS 1e-5f

__device__ __forceinline__ v8f wmma4(v2f a, v2f b, v8f c) {
  // D = A(16x4 f32) * B(4x16 f32) + C(16x16 f32)
  return __builtin_amdgcn_wmma_f32_16x16x4_f32(false, a, false, b, (short)0, c, false, false);
}

// ------------------------------------------------------------------------
// Generic implicit-GEMM conv via WMMA f32 16x16x4, specialized on kernel
// size KS so the k -> (ic,ky,kx) decode uses constant-divisor arithmetic.
// in:(B,Cin,H,W)  w:(Cout,Cin,KS,KS)  out:(B,Cout,H,W)
// Optional bias / batchnorm (s,b,m,v) / relu fused in epilogue.
// block = 256 threads = 8 waves; each wave computes one 16x16 output tile.
// ------------------------------------------------------------------------
template <int KS>
__global__ __launch_bounds__(256)
void conv_wmma(const float* __restrict__ in, const float* __restrict__ w,
               const float* __restrict__ bias,
               const float* __restrict__ bn_s, const float* __restrict__ bn_b,
               const float* __restrict__ bn_m, const float* __restrict__ bn_v,
               float* __restrict__ out,
               int B, int Cin, int H, int W, int Cout, int relu)
{
  constexpr int KSQ = KS * KS;
  constexpr int PAD = KS / 2;
  const int N     = B * H * W;
  const int K     = Cin * KSQ;
  const int Kpad  = (K + 3) & ~3;
  const int Ntiles = (N + 15) >> 4;
  const int Mtiles = (Cout + 15) >> 4;
  const int totalJobs = Ntiles * Mtiles;

  const int wave = threadIdx.x >> 5;
  const int lane = threadIdx.x & 31;
  const int half = lane >> 4;
  const int lm   = lane & 15;

  const int job = blockIdx.x * (blockDim.x >> 5) + wave;
  if (job >= totalJobs) return;        // wave-uniform branch, EXEC stays all-1

  const int mt = job / Ntiles;
  const int nt = job - mt * Ntiles;
  const int m0 = mt << 4, n0 = nt << 4;

  const int n  = n0 + lm;              // output pixel for B/C fragments
  const bool nOK = (n < N);
  const int HW = H * W;
  int b = 0, y = 0, x = 0;
  if (nOK) { b = n / HW; int r = n - b * HW; y = r / W; x = r - y * W; }

  const int mA  = m0 + lm;             // output channel row for A fragment
  const bool mOK = (mA < Cout);
  const float* wrow = w + (size_t)mA * K;
  const float* inb  = in + (size_t)b * Cin * HW;

  v8f acc = {0.f,0.f,0.f,0.f,0.f,0.f,0.f,0.f};

  for (int k0 = 0; k0 < Kpad; k0 += 4) {
    const int kk = k0 + half * 2;
    __builtin_prefetch(wrow + kk + 32, 0, 0);   // global_prefetch_b8
    v2f a;
    a.x = (mOK && kk     < K) ? wrow[kk]     : 0.f;
    a.y = (mOK && kk + 1 < K) ? wrow[kk + 1] : 0.f;
    v2f bb;
#pragma unroll
    for (int t = 0; t < 2; ++t) {
      const int k = kk + t;
      float v = 0.f;
      if (k < K && nOK) {
        const int ic = k / KSQ;             // constant divisor
        const int r  = k - ic * KSQ;
        const int ky = r / KS;
        const int kx = r - ky * KS;
        const int iy = y + ky - PAD, ix = x + kx - PAD;
        if ((unsigned)iy < (unsigned)H && (unsigned)ix < (unsigned)W)
          v = inb[(ic * H + iy) * W + ix];
      }
      if (t == 0) bb.x = v; else bb.y = v;
    }
    acc = wmma4(a, bb, acc);
  }

#pragma unroll
  for (int r = 0; r < 8; ++r) {
    const int m = m0 + r + half * 8;   // C/D layout: vgpr r -> M=r (lanes 0-15), M=8+r (16-31)
    if (m < Cout && nOK) {
      float v = acc[r];
      if (bias) v += bias[m];
      if (bn_s) v = (v - bn_m[m]) * (bn_s[m] * rsqrtf(bn_v[m] + BN_EPS)) + bn_b[m];
      if (relu) v = fmaxf(v, 0.f);
      out[((b * Cout + m) * H + y) * W + x] = v;
    }
  }
}

// ------------------------------------------------------------------------
// Pre-transform one LCA dictionary into WMMA A-fragment layout in global
// scratch, k-pair interleaved so a lane's (k, k+1) pair is ONE b64 load:
//   flat index (k,m) -> (k>>1)*(MP*2) + m*2 + (k&1)
//   Wf holds w[m][ic][ky][kx]        (k = ic*25+ky*5+kx, MP = CoutPad)
//   Wt holds w[oc][m][4-ky][4-kx]    (k = oc*25+ky*5+kx, MP = CinPad)
// Zero-padded to WfK/WtK (multiple of 4) and CinPad/CoutPad (multiple of 16).
// ------------------------------------------------------------------------
__global__ void prep_lca_weights(const float* __restrict__ w,
                                 float* __restrict__ Wf, float* __restrict__ Wt,
                                 int Cin, int Cout, int WfK, int WtK,
                                 int CinPad, int CoutPad)
{
  const int nf = WfK * CoutPad;
  const int nt = WtK * CinPad;
  for (int i = blockIdx.x * blockDim.x + threadIdx.x; i < nf + nt;
       i += gridDim.x * blockDim.x) {
    if (i < nf) {
      const int k = i / CoutPad, m = i - k * CoutPad;
      float v = 0.f;
      if (k < Cin * 25 && m < Cout) {
        const int ic = k / 25, r = k - ic * 25, ky = r / 5, kx = r - (r / 5) * 5;
        v = w[((m * Cin + ic) * 5 + ky) * 5 + kx];
      }
      Wf[(k >> 1) * (CoutPad * 2) + m * 2 + (k & 1)] = v;
    } else {
      const int j = i - nf;
      const int k = j / CinPad, m = j - k * CinPad;
      float v = 0.f;
      if (k < Cout * 25 && m < Cin) {
        const int oc = k / 25, r = k - oc * 25, ky = r / 5, kx = r - (r / 5) * 5;
        v = w[((oc * Cin + m) * 5 + (4 - ky)) * 5 + (4 - kx)];
      }
      Wt[(k >> 1) * (CinPad * 2) + m * 2 + (k & 1)] = v;
    }
  }
}

// ------------------------------------------------------------------------
// Persistent LCA sparse-coding kernel. One workgroup per batch image.
// State lives in LDS with a 2-pixel zero halo (36x36 per channel) so the
// inner gather loop is completely unconditional:
//   B-frag value = relu(u_pad[offs[k] + pbase] - lam)
// offs[] tables are iteration-invariant and built once in LDS.
// A fragments come from the pre-transformed pair-interleaved arrays
// (one global_load_b64 per chunk, L0/L2-hot).
// Per iteration (2 barriers):
//   A: err = x - conv(relu(u-lam), W^T)
//   B: u  += (conv(err, W) + relu(u-lam) - u) / tau
// Final: out = batchnorm(relu(u - lam)).
// ------------------------------------------------------------------------
__global__ __launch_bounds__(256)
void lca_wmma(const float* __restrict__ x_g,
              const float* __restrict__ WfG, const float* __restrict__ WtG,
              const float* __restrict__ bn_s, const float* __restrict__ bn_b,
              const float* __restrict__ bn_m, const float* __restrict__ bn_v,
              float* __restrict__ out_g,
              int Cin, int Cout, int n_iters, float tau_inv, float lam)
{
  const int HW    = 1024;            // 32x32 image
  const int PHW   = 1296;            // 36x36 padded channel
  const int bimg  = blockIdx.x;
  const int tid   = threadIdx.x;
  const int wave  = tid >> 5, lane = tid & 31, half = lane >> 4, lm = lane & 15;
  const int NW    = blockDim.x >> 5;

  const int CinPad  = (Cin  + 15) & ~15;
  const int CoutPad = (Cout + 15) & ~15;
  const int WfK = (Cin  * 25 + 3) & ~3;   // K of forward conv (padded to 4)
  const int WtK = (Cout * 25 + 3) & ~3;   // K of transpose conv

  extern __shared__ float sm[];
  float* u_pad   = sm;                       // Cout*1296 (zero halo)
  float* err_pad = u_pad + Cout * PHW;       // Cin *1296 (zero halo)
  int*   uoffs   = (int*)(err_pad + Cin * PHW); // WtK gather offsets into u_pad
  int*   eoffs   = uoffs + WtK;                 // WfK gather offsets into err_pad

  // ---- init: zero padded state, build iteration-invariant gather tables
  for (int i = tid; i < (Cout + Cin) * PHW; i += blockDim.x) u_pad[i] = 0.f;
  for (int i = tid; i < WtK; i += blockDim.x) {
    int off = 0;
    if (i < Cout * 25) {
      const int oc = i / 25, r = i - oc * 25, ky = r / 5, kx = r - (r / 5) * 5;
      off = oc * PHW + ky * 36 + kx;
    }
    uoffs[i] = off;                 // pad-k -> offset 0 (in-bounds read; its A weight is 0)
  }
  for (int i = tid; i < WfK; i += blockDim.x) {
    int off = 0;
    if (i < Cin * 25) {
      const int ic = i / 25, r = i - ic * 25, ky = r / 5, kx = r - (r / 5) * 5;
      off = ic * PHW + ky * 36 + kx;
    }
    eoffs[i] = off;
  }
  __syncthreads();

  const int jobsA = (CinPad  >> 4) * 64;  // M-tiles x N-tiles (1024 px / 16)
  const int jobsB = (CoutPad >> 4) * 64;
  const float* xb = x_g + (size_t)bimg * Cin * HW;

  for (int it = 0; it < n_iters; ++it) {
    // ---------------- step A: err = x - Phi a --------------------------
    for (int job = wave; job < jobsA; job += NW) {
      const int mt = job >> 6, nt = job & 63;
      const int m0 = mt << 4, n0 = nt << 4;
      const int p = n0 + lm, y = p >> 5, xx = p & 31;
      const int pbase = y * 36 + xx;          // padded-coords gather base
      const v2f* wA = (const v2f*)WtG + (m0 + lm);   // pair-interleaved [k/2][m]
      v8f acc = {0.f,0.f,0.f,0.f,0.f,0.f,0.f,0.f};
#pragma unroll 4
      for (int k0 = 0; k0 < WtK; k0 += 4) {
        const int kk = k0 + half * 2;
        const v2f a = wA[(kk >> 1) * CinPad];        // one global_load_b64
        v2f bb;
        bb.x = fmaxf(u_pad[uoffs[kk]     + pbase] - lam, 0.f);
        bb.y = fmaxf(u_pad[uoffs[kk + 1] + pbase] - lam, 0.f);
        acc = wmma4(a, bb, acc);
      }
#pragma unroll
      for (int r = 0; r < 8; ++r) {
        const int m = m0 + r + half * 8;
        if (m < Cin) err_pad[m * PHW + pbase + 74] = xb[m * HW + p] - acc[r];
      }
    }
    __syncthreads();
    // ---------------- step B: u += (Phi^T err + a - u)/tau -------------
    for (int job = wave; job < jobsB; job += NW) {
      const int mt = job >> 6, nt = job & 63;
      const int m0 = mt << 4, n0 = nt << 4;
      const int p = n0 + lm, y = p >> 5, xx = p & 31;
      const int pbase = y * 36 + xx;
      const v2f* wA = (const v2f*)WfG + (m0 + lm);
      v8f acc = {0.f,0.f,0.f,0.f,0.f,0.f,0.f,0.f};
#pragma unroll 4
      for (int k0 = 0; k0 < WfK; k0 += 4) {
        const int kk = k0 + half * 2;
        const v2f a = wA[(kk >> 1) * CoutPad];
        v2f bb;
        bb.x = err_pad[eoffs[kk]     + pbase];
        bb.y = err_pad[eoffs[kk + 1] + pbase];
        acc = wmma4(a, bb, acc);
      }
#pragma unroll
      for (int r = 0; r < 8; ++r) {
        const int m = m0 + r + half * 8;
        if (m < Cout) {
          const int idx = m * PHW + pbase + 74;   // (y+2)*36 + (x+2)
          const float uo = u_pad[idx];
          const float av = fmaxf(uo - lam, 0.f);
          u_pad[idx] = uo + (acc[r] + av - uo) * tau_inv;
        }
      }
    }
    __syncthreads();
  }

  // ---- output: batchnorm(relu(u - lam))  (relu of nonneg a is identity)
  for (int i = tid; i < Cout * HW; i += blockDim.x) {
    const int c = i >> 10, p = i & 1023;
    const float a   = fmaxf(u_pad[c * PHW + ((p >> 5) + 2) * 36 + (p & 31) + 2] - lam, 0.f);
    const float inv = bn_s[c] * rsqrtf(bn_v[c] + BN_EPS);
    out_g[((size_t)bimg * Cout + c) * HW + p] = (a - bn_m[c]) * inv + bn_b[c];
  }
}

// ------------------------------------------------------------------------
__global__ void maxpool2_kernel(const float* __restrict__ in, float* __restrict__ out,
                                int B, int C, int H, int W)
{
  const int Ho = H >> 1, Wo = W >> 1;
  const int total = B * C * Ho * Wo;
  const int i = blockIdx.x * blockDim.x + threadIdx.x;
  if (i >= total) return;
  const int wo = i % Wo; int t = i / Wo;
  const int ho = t % Ho; t /= Ho;
  const int c = t % C; const int b = t / C;
  const float* p = in + (((size_t)b * C + c) * H + ho * 2) * W + wo * 2;
  out[i] = fmaxf(fmaxf(p[0], p[1]), fmaxf(p[W], p[W + 1]));
}

__global__ void adapool_kernel(const float* __restrict__ in, float* __restrict__ out,
                               int B, int C, int H, int W, int O)
{
  const int total = B * C * O * O;
  const int i = blockIdx.x * blockDim.x + threadIdx.x;
  if (i >= total) return;
  const int j = i % O; int t = i / O;
  const int ii = t % O; t /= O;
  const int c = t % C; const int b = t / C;
  const int h0 = (ii * H) / O, h1 = ((ii + 1) * H + O - 1) / O;
  const int w0 = (j * W) / O,  w1 = ((j + 1) * W + O - 1) / O;
  float s = 0.f; int cnt = 0;
  for (int y = h0; y < h1; ++y)
    for (int x = w0; x < w1; ++x) { s += in[(((size_t)b * C + c) * H + y) * W + x]; ++cnt; }
  out[i] = s / (float)cnt;
}

// ------------------------------------------------------------------------
static inline void launch_conv(const float* in, const float* w, const float* bias,
                               const float* s, const float* bb, const float* m, const float* v,
                               float* out, int B, int Cin, int H, int W, int Cout,
                               int ks, int relu, hipStream_t stream)
{
  const int N = B * H * W;
  const int jobs = ((N + 15) / 16) * ((Cout + 15) / 16);
  const int blocks = (jobs + 7) / 8;   // 8 waves per 256-thread block
  if (ks == 3)
    conv_wmma<3><<<blocks, 256, 0, stream>>>(in, w, bias, s, bb, m, v, out,
                                             B, Cin, H, W, Cout, relu);
  else
    conv_wmma<1><<<blocks, 256, 0, stream>>>(in, w, bias, s, bb, m, v, out,
                                             B, Cin, H, W, Cout, relu);
}

extern "C" void kernel_launch(void* const* d_in, const int* in_sizes, int n_in,
                              void* d_out, int out_size, void* d_ws, size_t ws_size,
                              hipStream_t stream) {
  (void)in_sizes; (void)n_in; (void)out_size; (void)ws_size;
  // setup_inputs() flattening order:
  // 0:x 1:lca1_w 2:lca2_w 3-6:bn1(s,b,m,v) 7-10:bn2 11:head_w 12:head_b
  // 13-25:vgg_w 26-38:vgg_b 39-90:vgg_bn(13x4) 91:down_w 92:down_b
  // 93:fc1_w 94:fc1_b 95:fc2_w 96:fc2_b
  const float* x      = (const float*)d_in[0];
  const float* lca1_w = (const float*)d_in[1];
  const float* lca2_w = (const float*)d_in[2];
  const float* head_w = (const float*)d_in[11];
  const float* head_b = (const float*)d_in[12];
  const float* down_w = (const float*)d_in[91];
  const float* down_b = (const float*)d_in[92];
  const float* fc1_w  = (const float*)d_in[93];
  const float* fc1_b  = (const float*)d_in[94];
  const float* fc2_w  = (const float*)d_in[95];
  const float* fc2_b  = (const float*)d_in[96];

  float* bufA = (float*)d_ws;
  float* bufB = bufA + (size_t)8 * 64 * 32 * 32;   // two 2 MB ping-pong buffers

  // LCA dictionary scratch (pair-interleaved A-fragment layout)
  // layer1: Cin=3,Cout=16  -> WfK1=76, WtK1=400, CinPad1=16, CoutPad1=16
  // layer2: Cin=16,Cout=32 -> WfK2=400,WtK2=800, CinPad2=16, CoutPad2=32
  float* Wf1 = bufB + (size_t)8 * 64 * 32 * 32;
  float* Wt1 = Wf1 + 76 * 16;
  float* Wf2 = Wt1 + 400 * 16;
  float* Wt2 = Wf2 + 400 * 32;

  const float TAU_INV = 1.0f / 1000.0f;
  const float LAM     = 0.5f;
  const int   ITERS   = 500;
  const int   PHW     = 1296;   // 36x36 padded channel plane

  // ---- stage LCA dictionaries into WMMA A-fragment layout ----
  prep_lca_weights<<<8, 256, 0, stream>>>(lca1_w, Wf1, Wt1, 3, 16, 76, 400, 16, 16);
  prep_lca_weights<<<32, 256, 0, stream>>>(lca2_w, Wf2, Wt2, 16, 32, 400, 800, 16, 32);

  // ---- LCA layer 1: 3 -> 16, output already batchnormed -> bufA
  {
    const size_t smem = (size_t)((16 + 3) * PHW + 400 + 76) * sizeof(float);
    lca_wmma<<<8, 256, smem, stream>>>(x, Wf1, Wt1,
        (const float*)d_in[3], (const float*)d_in[4], (const float*)d_in[5], (const float*)d_in[6],
        bufA, 3, 16, ITERS, TAU_INV, LAM);
  }
  // ---- LCA layer 2: 16 -> 32 -> bufB   (~249 KB LDS, fits 320 KB WGP)
  {
    const size_t smem = (size_t)((32 + 16) * PHW + 800 + 400) * sizeof(float);
    lca_wmma<<<8, 256, smem, stream>>>(bufA, Wf2, Wt2,
        (const float*)d_in[7], (const float*)d_in[8], (const float*)d_in[9], (const float*)d_in[10],
        bufB, 16, 32, ITERS, TAU_INV, LAM);
  }
  // ---- head 1x1 conv: 32 -> 3 (bias only) -> bufA
  launch_conv(bufB, head_w, head_b, nullptr, nullptr, nullptr, nullptr,
              bufA, 8, 32, 32, 32, 3, 1, 0, stream);

  // ---- VGG16-bn features ----
  static const int cfg[18] = {64,64,-1,128,128,-1,256,256,256,-1,512,512,512,-1,512,512,512,-1};
  float* cur = bufA; float* nxt = bufB;
  int Cin = 3, H = 32, W = 32, ci = 0;
  for (int e = 0; e < 18; ++e) {
    if (cfg[e] < 0) {
      const int total = 8 * Cin * (H / 2) * (W / 2);
      maxpool2_kernel<<<(total + 255) / 256, 256, 0, stream>>>(cur, nxt, 8, Cin, H, W);
      H /= 2; W /= 2;
    } else {
      const int Cout = cfg[e];
      launch_conv(cur, (const float*)d_in[13 + ci], (const float*)d_in[26 + ci],
                  (const float*)d_in[39 + 4 * ci + 0], (const float*)d_in[39 + 4 * ci + 1],
                  (const float*)d_in[39 + 4 * ci + 2], (const float*)d_in[39 + 4 * ci + 3],
                  nxt, 8, Cin, H, W, Cout, 3, 1, stream);
      Cin = Cout; ++ci;
    }
    float* tmp = cur; cur = nxt; nxt = tmp;
  }
  // cur: (8,512,1,1)

  // ---- adaptive avg pool to 2x2 (replicates the 1x1 feature) ----
  {
    const int total = 8 * 512 * 2 * 2;
    adapool_kernel<<<(total + 255) / 256, 256, 0, stream>>>(cur, nxt, 8, 512, H, W, 2);
    float* tmp = cur; cur = nxt; nxt = tmp;
  }
  // ---- down 1x1 conv: 512 -> 256 at 2x2 -> flat (8,1024) ----
  launch_conv(cur, down_w, down_b, nullptr, nullptr, nullptr, nullptr,
              nxt, 8, 512, 2, 2, 256, 1, 0, stream);
  { float* tmp = cur; cur = nxt; nxt = tmp; }
  // ---- fc1: (8,1024) @ (512,1024)^T + b, relu  (as 1x1 conv) ----
  launch_conv(cur, fc1_w, fc1_b, nullptr, nullptr, nullptr, nullptr,
              nxt, 8, 1024, 1, 1, 512, 1, 1, stream);
  { float* tmp = cur; cur = nxt; nxt = tmp; }
  // ---- fc2: (8,512) @ (10,512)^T + b -> d_out (8,10) ----
  launch_conv(cur, fc2_w, fc2_b, nullptr, nullptr, nullptr, nullptr,
              (float*)d_out, 8, 512, 1, 1, 10, 1, 0, stream);
}